// QuConVXYZ_1297080123413
// MI455X (gfx1250) — compile-verified
//
#include <hip/hip_runtime.h>
#include <hip/hip_bf16.h>

// ---------------------------------------------------------------------------
// QuConv quantum layer for MI455X (gfx1250).
//   out[b,i,j] = sum_p y[16p+i] * conj(y[16p+j]),  y = U x[b]
// Kernel 1: build 256x256 complex unitary U (planar re/im) in d_ws.
// Kernel 2: fp32 WMMA GEMM (Y = X * U^T, re & im) fused with partial trace.
// ---------------------------------------------------------------------------

typedef float v2f __attribute__((ext_vector_type(2)));
typedef float v8f __attribute__((ext_vector_type(8)));

#define DIM   256
#define WMUL  0.63245553203367586640f   // sqrt(2)/sqrt(5)

struct cplx { float x, y; };

__device__ __forceinline__ cplx cmul(const cplx a, const cplx b) {
    return cplx{ fmaf(a.x, b.x, -a.y * b.y), fmaf(a.x, b.y, a.y * b.x) };
}
__device__ __forceinline__ cplx cadd(const cplx a, const cplx b) {
    return cplx{ a.x + b.x, a.y + b.y };
}

__device__ __forceinline__ void mm2(cplx C[2][2], const cplx A[2][2], const cplx B[2][2]) {
    for (int i = 0; i < 2; ++i)
        for (int j = 0; j < 2; ++j) {
            cplx s{0.f, 0.f};
            for (int k = 0; k < 2; ++k) s = cadd(s, cmul(A[i][k], B[k][j]));
            C[i][j] = s;
        }
}
__device__ __forceinline__ void mm4(cplx C[4][4], const cplx A[4][4], const cplx B[4][4]) {
    for (int i = 0; i < 4; ++i)
        for (int j = 0; j < 4; ++j) {
            cplx s{0.f, 0.f};
            for (int k = 0; k < 4; ++k) s = cadd(s, cmul(A[i][k], B[k][j]));
            C[i][j] = s;
        }
}

// Apply fused single-qubit gate G = Rz(t3)*Ry(t2)*Rx(t1) on qubit q (bit 7-q).
__device__ void apply1q(int t, int q, float t1, float t2, float t3,
                        float* __restrict__ Ure, float* __restrict__ Uim) {
    const float h1 = 0.5f * t1, h2 = 0.5f * t2, h3 = 0.5f * t3;
    const float c1 = cosf(h1), s1 = sinf(h1);
    const float c2 = cosf(h2), s2 = sinf(h2);
    const float c3 = cosf(h3), s3 = sinf(h3);
    const cplx Rx[2][2] = { { {c1, 0.f}, {0.f, -s1} }, { {0.f, -s1}, {c1, 0.f} } };
    const cplx Ry[2][2] = { { {c2, 0.f}, {-s2, 0.f} }, { {s2, 0.f}, {c2, 0.f} } };
    const cplx Rz[2][2] = { { {c3, -s3}, {0.f, 0.f} }, { {0.f, 0.f}, {c3, s3} } };
    cplx T[2][2], G[2][2];
    mm2(T, Ry, Rx);
    mm2(G, Rz, T);

    const int p = 7 - q;
    const int bit = 1 << p;
    for (int idx = t; idx < 128 * DIM; idx += 256) {
        const int pr = idx >> 8;          // pair index (row with bit p removed)
        const int c  = idx & 255;         // column
        const int r0 = ((pr >> p) << (p + 1)) | (pr & (bit - 1));
        const int i0 = r0 * DIM + c;
        const int i1 = i0 + bit * DIM;
        const cplx u0{Ure[i0], Uim[i0]};
        const cplx u1{Ure[i1], Uim[i1]};
        const cplx n0 = cadd(cmul(G[0][0], u0), cmul(G[0][1], u1));
        const cplx n1 = cadd(cmul(G[1][0], u0), cmul(G[1][1], u1));
        Ure[i0] = n0.x; Uim[i0] = n0.y;
        Ure[i1] = n1.x; Uim[i1] = n1.y;
    }
    __threadfence_block();
    __syncthreads();
}

// Apply fused two-qubit gate G = Rzz(t3)*Ryy(t2)*Rxx(t1) on qubits (q,q+1),
// i.e. row bits (7-q, 6-q); local index L = 2a + b with a at the higher bit.
__device__ void apply2q(int t, int q, float t1, float t2, float t3,
                        float* __restrict__ Ure, float* __restrict__ Uim) {
    const float h1 = 0.5f * t1, h2 = 0.5f * t2, h3 = 0.5f * t3;
    const float c1 = cosf(h1), s1 = sinf(h1);
    const float c2 = cosf(h2), s2 = sinf(h2);
    const float c3 = cosf(h3), s3 = sinf(h3);

    cplx Gxx[4][4] = {};
    cplx Gyy[4][4] = {};
    cplx Gzz[4][4] = {};
    for (int i = 0; i < 4; ++i) {
        Gxx[i][i]     = cplx{c1, 0.f};
        Gxx[i][3 - i] = cplx{0.f, -s1};
        Gyy[i][i]     = cplx{c2, 0.f};
    }
    Gyy[0][3] = cplx{0.f,  s2};
    Gyy[1][2] = cplx{0.f, -s2};
    Gyy[2][1] = cplx{0.f, -s2};
    Gyy[3][0] = cplx{0.f,  s2};
    Gzz[0][0] = cplx{c3, -s3};
    Gzz[1][1] = cplx{c3,  s3};
    Gzz[2][2] = cplx{c3,  s3};
    Gzz[3][3] = cplx{c3, -s3};

    cplx T[4][4], G[4][4];
    mm4(T, Gyy, Gxx);
    mm4(G, Gzz, T);

    const int p0 = 6 - q;                 // lower bit of the adjacent pair
    const int b0 = 1 << p0;
    const int b1 = b0 << 1;               // higher bit (7-q)
    const int off[4] = {0, b0 * DIM, b1 * DIM, (b1 | b0) * DIM};  // L = 2a+b

    for (int idx = t; idx < 64 * DIM; idx += 256) {
        const int qd = idx >> 8;          // row with bits p0,p0+1 removed
        const int c  = idx & 255;
        const int r00  = ((qd >> p0) << (p0 + 2)) | (qd & (b0 - 1));
        const int base = r00 * DIM + c;
        cplx u[4];
        #pragma unroll
        for (int L = 0; L < 4; ++L) u[L] = cplx{Ure[base + off[L]], Uim[base + off[L]]};
        #pragma unroll
        for (int L = 0; L < 4; ++L) {
            cplx s{0.f, 0.f};
            #pragma unroll
            for (int M = 0; M < 4; ++M) s = cadd(s, cmul(G[L][M], u[M]));
            Ure[base + off[L]] = s.x;
            Uim[base + off[L]] = s.y;
        }
    }
    __threadfence_block();
    __syncthreads();
}

__global__ __launch_bounds__(256) void build_unitary_kernel(
    const float* __restrict__ w, float* __restrict__ Ure, float* __restrict__ Uim) {
    const int t = threadIdx.x;
    // U = identity
    for (int idx = t; idx < DIM * DIM; idx += 256) {
        Ure[idx] = ((idx >> 8) == (idx & 255)) ? 1.0f : 0.0f;
        Uim[idx] = 0.0f;
    }
    __threadfence_block();
    __syncthreads();

    // first XYZ layer
    for (int q = 0; q < 8; ++q)
        apply1q(t, q, w[3*q] * WMUL, w[3*q+1] * WMUL, w[3*q+2] * WMUL, Ure, Uim);
    // entangling ladder
    for (int q = 0; q < 7; ++q) {
        const int i = 24 + 3 * q;
        apply2q(t, q, w[i] * WMUL, w[i+1] * WMUL, w[i+2] * WMUL, Ure, Uim);
    }
    // second XYZ layer
    for (int q = 0; q < 8; ++q) {
        const int i = 45 + 3 * q;
        apply1q(t, q, w[i] * WMUL, w[i+1] * WMUL, w[i+2] * WMUL, Ure, Uim);
    }
}

// ---------------------------------------------------------------------------
// Fused GEMM + partial trace.
//   Block: 512 threads = 16 waves; block handles 16 batches, wave w handles
//   U rows [16w, 16w+16). fp32 WMMA 16x16x4: M=batch, N=U-row, K=column.
//   A(v,lane) = x[b0 + lane%16][k0 + 2*(lane/16) + v]
//   B(v,lane) = U[r0 + lane%16][k0 + 2*(lane/16) + v]      (B[k][n] = U[n][k])
//   D(v,lane) = Y[b0 + v + 8*(lane/16)][r0 + lane%16]
// ---------------------------------------------------------------------------
__global__ __launch_bounds__(512) void qgemm_trace_kernel(
    const float* __restrict__ x,
    const float* __restrict__ Ure, const float* __restrict__ Uim,
    float* __restrict__ out) {
    __shared__ float ly_re[16][DIM];
    __shared__ float ly_im[16][DIM];

    const int tid  = threadIdx.x;
    const int wave = tid >> 5;
    const int lane = tid & 31;
    const int b0   = blockIdx.x * 16;
    const int r0   = wave * 16;
    const int m    = lane & 15;
    const int kh   = lane >> 4;

    const float* xrow = x   + (size_t)(b0 + m) * DIM + 2 * kh;
    const float* urr  = Ure + (size_t)(r0 + m) * DIM + 2 * kh;
    const float* uri  = Uim + (size_t)(r0 + m) * DIM + 2 * kh;

    v8f cre = {};
    v8f cim = {};
    #pragma unroll 4
    for (int k0 = 0; k0 < DIM; k0 += 4) {
        const v2f a  = *(const v2f*)(xrow + k0);
        const v2f br = *(const v2f*)(urr + k0);
        const v2f bi = *(const v2f*)(uri + k0);
        cre = __builtin_amdgcn_wmma_f32_16x16x4_f32(
                  false, a, false, br, (short)0, cre, false, false);
        cim = __builtin_amdgcn_wmma_f32_16x16x4_f32(
                  false, a, false, bi, (short)0, cim, false, false);
    }

    // D-matrix layout -> LDS:  row m = v + 8*(lane/16), col n = lane&15
    {
        const int n  = lane & 15;
        const int mh = lane >> 4;
        #pragma unroll
        for (int v = 0; v < 8; ++v) {
            const int mb = v + mh * 8;
            ly_re[mb][r0 + n] = cre[v];
            ly_im[mb][r0 + n] = cim[v];
        }
    }
    __syncthreads();

    // Partial trace over the high 4 qubits:
    //   out[b, i, j] = sum_p y[16p+i] * conj(y[16p+j])
    for (int idx = tid; idx < 16 * 256; idx += 512) {
        const int mb = idx >> 8;
        const int ij = idx & 255;
        const int i  = ij >> 4;
        const int j  = ij & 15;
        float sr = 0.f, si = 0.f;
        #pragma unroll
        for (int p = 0; p < 16; ++p) {
            const float ar = ly_re[mb][p * 16 + i];
            const float ai = ly_im[mb][p * 16 + i];
            const float br = ly_re[mb][p * 16 + j];
            const float bi = ly_im[mb][p * 16 + j];
            sr = fmaf(ar, br, fmaf(ai, bi, sr));   // Re(a * conj(b))
            si = fmaf(ai, br, si) - ar * bi;       // Im(a * conj(b))
        }
        const size_t o = ((size_t)(b0 + mb) * 256 + ij) * 2;  // complex64 interleaved
        out[o]     = sr;
        out[o + 1] = si;
    }
}

extern "C" void kernel_launch(void* const* d_in, const int* in_sizes, int n_in,
                              void* d_out, int out_size, void* d_ws, size_t ws_size,
                              hipStream_t stream) {
    const float* x = (const float*)d_in[0];   // [512,1,256] fp32
    const float* w = (const float*)d_in[1];   // [69] fp32
    // d_in[2] = dimA (== 4), baked into the kernel (dA = dB = 16).

    float* Ure = (float*)d_ws;                // 256*256 floats
    float* Uim = Ure + DIM * DIM;             // 256*256 floats (512 KB total)

    build_unitary_kernel<<<1, 256, 0, stream>>>(w, Ure, Uim);
    qgemm_trace_kernel<<<32, 512, 0, stream>>>(x, Ure, Uim, (float*)d_out);
}